// TextAttValMaskedMultiheadSelfAttention_21492016349443
// MI455X (gfx1250) — compile-verified
//
#include <hip/hip_runtime.h>
#include <math.h>

typedef float v2f __attribute__((ext_vector_type(2)));
typedef float v4f __attribute__((ext_vector_type(4)));
typedef float v8f __attribute__((ext_vector_type(8)));
typedef int   i4v __attribute__((ext_vector_type(4)));

#define BB 2
#define TT 2048
#define EE 1024
#define HH 16
#define DD 64

#if defined(__has_builtin)
#if __has_builtin(__builtin_amdgcn_global_load_async_to_lds_b128) && \
    __has_builtin(__builtin_amdgcn_s_wait_asynccnt)
#define USE_ASYNC_LDS 1
#endif
#endif

static __device__ __forceinline__ v8f wmma_f32(v2f a, v2f b, v8f c) {
  // V_WMMA_F32_16X16X4_F32: D = A(16x4) * B(4x16) + C(16x16)
  return __builtin_amdgcn_wmma_f32_16x16x4_f32(
      /*neg_a=*/false, a, /*neg_b=*/false, b,
      /*c_mod=*/(short)0, c, /*reuse_a=*/false, /*reuse_b=*/false);
}

// -----------------------------------------------------------------------------
// C[M x N] = A[M x K] @ W[N x K]^T + bias[N]
// 128-thread block (4 waves) computes a 64x64 tile. K chunks of 32 staged in
// LDS (padded stride 36 -> conflict-free 16-lane reads, 16B-aligned stores),
// double buffered so next chunk's async-to-LDS (or register-staged) loads
// overlap the current chunk's 32 WMMAs. B tile is shared by all 4 waves.
// -----------------------------------------------------------------------------
#define GEMM_LDA 36

__global__ __launch_bounds__(128) void gemm_xwT_bias_kernel(
    const float* __restrict__ A, const float* __restrict__ W,
    const float* __restrict__ bias, float* __restrict__ C,
    int M, int N, int K)
{
  __shared__ float lsA[2][64 * GEMM_LDA];
  __shared__ float lsB[2][64 * GEMM_LDA];

  const int t    = threadIdx.x;
  const int w    = t >> 5;
  const int lane = t & 31;
  const int half = lane >> 4;
  const int ln   = lane & 15;
  const int mb = blockIdx.x * 64;
  const int nb = blockIdx.y * 64;

  v8f acc0 = {}, acc1 = {}, acc2 = {}, acc3 = {};

#if !USE_ASYNC_LDS
  v4f rA[4], rB[4];
#endif

  // ---- stage chunk 0 ----
  {
    #pragma unroll
    for (int i = 0; i < 4; ++i) {
      const int idx = t + i * 128;       // 0..511
      const int row = idx >> 3;          // 0..63
      const int c4  = (idx & 7) << 2;    // 0,4,..,28
#if USE_ASYNC_LDS
      __builtin_amdgcn_global_load_async_to_lds_b128(
          (i4v*)(A + (size_t)(mb + row) * K + c4),
          (i4v*)&lsA[0][row * GEMM_LDA + c4], 0, 0);
      __builtin_amdgcn_global_load_async_to_lds_b128(
          (i4v*)(W + (size_t)(nb + row) * K + c4),
          (i4v*)&lsB[0][row * GEMM_LDA + c4], 0, 0);
#else
      rA[i] = *(const v4f*)(A + (size_t)(mb + row) * K + c4);
      rB[i] = *(const v4f*)(W + (size_t)(nb + row) * K + c4);
#endif
    }
#if USE_ASYNC_LDS
    __builtin_amdgcn_s_wait_asynccnt(0);
#else
    #pragma unroll
    for (int i = 0; i < 4; ++i) {
      const int idx = t + i * 128;
      const int row = idx >> 3;
      const int c4  = (idx & 7) << 2;
      *(v4f*)&lsA[0][row * GEMM_LDA + c4] = rA[i];
      *(v4f*)&lsB[0][row * GEMM_LDA + c4] = rB[i];
    }
#endif
    __syncthreads();
  }

  const int nchunks = K / 32;
  for (int c = 0; c < nchunks; ++c) {
    const int buf = c & 1;
    const int kc_next = (c + 1) * 32;

    // issue next chunk's loads before computing (overlap with WMMAs)
    if (c + 1 < nchunks) {
      #pragma unroll
      for (int i = 0; i < 4; ++i) {
        const int idx = t + i * 128;
        const int row = idx >> 3;
        const int c4  = (idx & 7) << 2;
#if USE_ASYNC_LDS
        __builtin_amdgcn_global_load_async_to_lds_b128(
            (i4v*)(A + (size_t)(mb + row) * K + kc_next + c4),
            (i4v*)&lsA[buf ^ 1][row * GEMM_LDA + c4], 0, 0);
        __builtin_amdgcn_global_load_async_to_lds_b128(
            (i4v*)(W + (size_t)(nb + row) * K + kc_next + c4),
            (i4v*)&lsB[buf ^ 1][row * GEMM_LDA + c4], 0, 0);
#else
        rA[i] = *(const v4f*)(A + (size_t)(mb + row) * K + kc_next + c4);
        rB[i] = *(const v4f*)(W + (size_t)(nb + row) * K + kc_next + c4);
#endif
      }
    }

    // ---- compute current chunk from LDS: 8 K-steps x 4 N-tiles ----
    const float* la = &lsA[buf][(16 * w + ln) * GEMM_LDA];
    const float* lb = &lsB[buf][ln * GEMM_LDA];
    #pragma unroll
    for (int ks = 0; ks < 8; ++ks) {
      const int ka = ks * 4 + 2 * half;
      v2f av = *(const v2f*)(la + ka);
      v2f b0 = *(const v2f*)(lb + ka);
      v2f b1 = *(const v2f*)(lb + 16 * GEMM_LDA + ka);
      v2f b2 = *(const v2f*)(lb + 32 * GEMM_LDA + ka);
      v2f b3 = *(const v2f*)(lb + 48 * GEMM_LDA + ka);
      acc0 = wmma_f32(av, b0, acc0);
      acc1 = wmma_f32(av, b1, acc1);
      acc2 = wmma_f32(av, b2, acc2);
      acc3 = wmma_f32(av, b3, acc3);
    }

    if (c + 1 < nchunks) {
#if USE_ASYNC_LDS
      __builtin_amdgcn_s_wait_asynccnt(0);
#else
      #pragma unroll
      for (int i = 0; i < 4; ++i) {
        const int idx = t + i * 128;
        const int row = idx >> 3;
        const int c4  = (idx & 7) << 2;
        *(v4f*)&lsA[buf ^ 1][row * GEMM_LDA + c4] = rA[i];
        *(v4f*)&lsB[buf ^ 1][row * GEMM_LDA + c4] = rB[i];
      }
#endif
      __syncthreads();
    }
  }

  // ---- epilogue: bias add + coalesced stores ----
  #pragma unroll
  for (int r = 0; r < 8; ++r) {
    const int row = mb + 16 * w + r + 8 * half;
    float* crow = C + (size_t)row * N + nb + ln;
    crow[ 0] = acc0[r] + bias[nb + ln +  0];
    crow[16] = acc1[r] + bias[nb + ln + 16];
    crow[32] = acc2[r] + bias[nb + ln + 32];
    crow[48] = acc3[r] + bias[nb + ln + 48];
  }
}

// -----------------------------------------------------------------------------
// xw[b,h,t,e] = sum_d x[b,t,h*D+d] * w_att[h,d,e]  (per-head 2048x64 @ 64x64)
// One wave per 16x64 strip; 2-stage register pipeline so the next K-step's
// loads are in flight during the current WMMAs.
// -----------------------------------------------------------------------------
__global__ __launch_bounds__(32) void head_xw_kernel(
    const float* __restrict__ x, const float* __restrict__ w_att,
    float* __restrict__ xw)
{
  const int lane = threadIdx.x;
  const int half = lane >> 4;
  const int ln   = lane & 15;
  const int tb = blockIdx.x * 16;
  const int bh = blockIdx.y;
  const int b = bh / HH, h = bh % HH;

  const float* xrow = x + ((size_t)b * TT + tb + ln) * EE + h * DD;
  const float* wh = w_att + (size_t)h * DD * DD;

  v8f acc0 = {}, acc1 = {}, acc2 = {}, acc3 = {};

  v2f av, b0, b1, b2, b3;
  {
    const int ka = 2 * half;
    av = *(const v2f*)(xrow + ka);
    b0.x = wh[ka * DD + ln +  0]; b0.y = wh[(ka + 1) * DD + ln +  0];
    b1.x = wh[ka * DD + ln + 16]; b1.y = wh[(ka + 1) * DD + ln + 16];
    b2.x = wh[ka * DD + ln + 32]; b2.y = wh[(ka + 1) * DD + ln + 32];
    b3.x = wh[ka * DD + ln + 48]; b3.y = wh[(ka + 1) * DD + ln + 48];
  }

  #pragma unroll
  for (int k0 = 0; k0 < DD; k0 += 4) {
    v2f avn = av, b0n = b0, b1n = b1, b2n = b2, b3n = b3;
    if (k0 + 4 < DD) {
      const int ka = k0 + 4 + 2 * half;
      avn = *(const v2f*)(xrow + ka);
      b0n.x = wh[ka * DD + ln +  0]; b0n.y = wh[(ka + 1) * DD + ln +  0];
      b1n.x = wh[ka * DD + ln + 16]; b1n.y = wh[(ka + 1) * DD + ln + 16];
      b2n.x = wh[ka * DD + ln + 32]; b2n.y = wh[(ka + 1) * DD + ln + 32];
      b3n.x = wh[ka * DD + ln + 48]; b3n.y = wh[(ka + 1) * DD + ln + 48];
    }
    acc0 = wmma_f32(av, b0, acc0);
    acc1 = wmma_f32(av, b1, acc1);
    acc2 = wmma_f32(av, b2, acc2);
    acc3 = wmma_f32(av, b3, acc3);
    av = avn; b0 = b0n; b1 = b1n; b2 = b2n; b3 = b3n;
  }

  #pragma unroll
  for (int r = 0; r < 8; ++r) {
    const size_t o = ((size_t)bh * TT + tb + r + 8 * half) * DD;
    xw[o + ln +  0] = acc0[r];
    xw[o + ln + 16] = acc1[r];
    xw[o + ln + 32] = acc2[r];
    xw[o + ln + 48] = acc3[r];
  }
}

// -----------------------------------------------------------------------------
// Fused flash attention per (b,h): for each 16-row query tile,
//   S = xw_tile(16x64) @ x_h^T   (online over 128 s-tiles of 16)
//   online softmax (no (T,T) materialization)
//   O += P @ vx_h_tile
// All 16 score-B fragments are staged into registers (one wait -> 16 WMMAs),
// and V fragments are issued BEFORE the softmax so their latency hides under
// the VALU/shuffle work. P is converted C-layout -> A-layout via a 1KB LDS
// bounce (single-wave block: __syncthreads degenerates to an LDS wait).
// -----------------------------------------------------------------------------
__global__ __launch_bounds__(32) void flash_attn_kernel(
    const float* __restrict__ x, const float* __restrict__ xw,
    const float* __restrict__ vx, float* __restrict__ o)
{
  __shared__ float pshm[16 * 16];
  const int lane = threadIdx.x;
  const int half = lane >> 4;
  const int ln   = lane & 15;
  const int tb = blockIdx.x * 16;
  const int bh = blockIdx.y;
  const int b = bh / HH, h = bh % HH;

  // Preload A fragments of the query tile: XW rows tb..tb+15, K = D = 64.
  v2f aq[16];
  {
    const float* qrow = xw + ((size_t)bh * TT + tb + ln) * DD;
    #pragma unroll
    for (int ks = 0; ks < 16; ++ks)
      aq[ks] = *(const v2f*)(qrow + ks * 4 + 2 * half);
  }

  v8f o0 = {}, o1 = {}, o2 = {}, o3 = {};
  float m[8], l[8];
  #pragma unroll
  for (int r = 0; r < 8; ++r) { m[r] = -INFINITY; l[r] = 0.0f; }

  const float* xh_base = x  + (size_t)b * TT * EE + h * DD;
  const float* v_base  = vx + (size_t)b * TT * EE + h * DD;
  const float scale = 0.125f; // 1/sqrt(64)

  for (int sb = 0; sb < TT; sb += 16) {
    // ---- stage score-B fragments (16 loads, then one chained WMMA run) ----
    v2f bs[16];
    const float* srow = xh_base + (size_t)(sb + ln) * EE;
    #pragma unroll
    for (int ks = 0; ks < 16; ++ks)
      bs[ks] = *(const v2f*)(srow + ks * 4 + 2 * half);

    // ---- issue V-tile fragment loads early (independent of softmax) ----
    v2f vb0[4], vb1[4], vb2[4], vb3[4];
    #pragma unroll
    for (int ks = 0; ks < 4; ++ks) {
      const int ka = ks * 4 + 2 * half;
      const float* vr0 = v_base + (size_t)(sb + ka) * EE;
      const float* vr1 = vr0 + EE;
      vb0[ks].x = vr0[ln +  0]; vb0[ks].y = vr1[ln +  0];
      vb1[ks].x = vr0[ln + 16]; vb1[ks].y = vr1[ln + 16];
      vb2[ks].x = vr0[ln + 32]; vb2[ks].y = vr1[ln + 32];
      vb3[ks].x = vr0[ln + 48]; vb3[ks].y = vr1[ln + 48];
    }

    // ---- scores: S = XW @ Xh_s^T (K=64) ----
    v8f S = {};
    #pragma unroll
    for (int ks = 0; ks < 16; ++ks)
      S = wmma_f32(aq[ks], bs[ks], S);

    // ---- online softmax per row (rows r+8*half; 16-lane-half reductions) ----
    float p[8];
    #pragma unroll
    for (int r = 0; r < 8; ++r) {
      float s = S[r] * scale;
      float mx = s;
      mx = fmaxf(mx, __shfl_xor(mx, 1));
      mx = fmaxf(mx, __shfl_xor(mx, 2));
      mx = fmaxf(mx, __shfl_xor(mx, 4));
      mx = fmaxf(mx, __shfl_xor(mx, 8));
      const float mnew  = fmaxf(m[r], mx);
      const float alpha = __expf(m[r] - mnew);
      const float pe    = __expf(s - mnew);
      float rs = pe;
      rs += __shfl_xor(rs, 1);
      rs += __shfl_xor(rs, 2);
      rs += __shfl_xor(rs, 4);
      rs += __shfl_xor(rs, 8);
      l[r] = l[r] * alpha + rs;
      m[r] = mnew;
      p[r] = pe;
      o0[r] *= alpha; o1[r] *= alpha; o2[r] *= alpha; o3[r] *= alpha;
    }

    // ---- P: C/D layout -> A layout via LDS bounce ----
    __syncthreads();
    #pragma unroll
    for (int r = 0; r < 8; ++r)
      pshm[(r + 8 * half) * 16 + ln] = p[r];
    __syncthreads();

    // ---- O += P(16x16) @ V(16x64), K=16 in 4 WMMA steps ----
    #pragma unroll
    for (int ks = 0; ks < 4; ++ks) {
      const int ka = ks * 4 + 2 * half;
      v2f pa;
      pa.x = pshm[ln * 16 + ka];
      pa.y = pshm[ln * 16 + ka + 1];
      o0 = wmma_f32(pa, vb0[ks], o0);
      o1 = wmma_f32(pa, vb1[ks], o1);
      o2 = wmma_f32(pa, vb2[ks], o2);
      o3 = wmma_f32(pa, vb3[ks], o3);
    }
    __syncthreads(); // protect pshm (WAR) before next tile
  }

  // ---- normalize and store into (B,T,E) head slice ----
  #pragma unroll
  for (int r = 0; r < 8; ++r) {
    const float inv = 1.0f / l[r];
    const size_t off = ((size_t)b * TT + tb + r + 8 * half) * EE + h * DD;
    o[off + ln +  0] = o0[r] * inv;
    o[off + ln + 16] = o1[r] * inv;
    o[off + ln + 32] = o2[r] * inv;
    o[off + ln + 48] = o3[r] * inv;
  }
}

extern "C" void kernel_launch(void* const* d_in, const int* in_sizes, int n_in,
                              void* d_out, int out_size, void* d_ws, size_t ws_size,
                              hipStream_t stream) {
  const float* x      = (const float*)d_in[0];
  const float* w_att  = (const float*)d_in[1];
  const float* wv_w   = (const float*)d_in[2];
  const float* wv_b   = (const float*)d_in[3];
  const float* out_w  = (const float*)d_in[4];
  const float* out_b  = (const float*)d_in[5];
  float* out = (float*)d_out;

  float* ws = (float*)d_ws;
  const size_t nBTE = (size_t)BB * TT * EE;
  float* vx  = ws;             // (B,T,E) value projection
  float* xw  = ws + nBTE;      // (B,H,T,D) bilinear-projected queries
  float* oin = ws + 2 * nBTE;  // (B,T,E) attention output pre out-proj

  // vx = x @ wv_w^T + wv_b
  gemm_xwT_bias_kernel<<<dim3((BB * TT) / 64, EE / 64), dim3(128), 0, stream>>>(
      x, wv_w, wv_b, vx, BB * TT, EE, EE);

  // xw = x_h @ W_h per head
  head_xw_kernel<<<dim3(TT / 16, BB * HH), dim3(32), 0, stream>>>(x, w_att, xw);

  // fused scores + softmax + value aggregation
  flash_attn_kernel<<<dim3(TT / 16, BB * HH), dim3(32), 0, stream>>>(x, xw, vx, oin);

  // out = o @ out_w^T + out_b
  gemm_xwT_bias_kernel<<<dim3((BB * TT) / 64, EE / 64), dim3(128), 0, stream>>>(
      oin, out_w, out_b, out, BB * TT, EE, EE);
}